// FoveatedMnist_40905268527290
// MI455X (gfx1250) — compile-verified
//
#include <hip/hip_runtime.h>
#include <hip/hip_bf16.h>
#include <math.h>
#include <stdint.h>

// ---------------- constants ----------------
#define NIMG   1024
#define HH     224
#define WW     224
#define TT     16
#define PP     8
#define DIM    384
#define MLPD   768
#define DEPTH  4
#define NFREQS 64
#define HD     64
#define NHEADS 6
#define NT     (NIMG * TT)      // 16384
#define LPATCH 284
#define ISZ    (225 * 225)      // padded integral image per frame

typedef __attribute__((ext_vector_type(2))) float v2f;
typedef __attribute__((ext_vector_type(8))) float v8f;

__device__ __forceinline__ v8f wmma4(v2f a, v2f b, v8f c) {
  // D = A(16x4,f32) * B(4x16,f32) + C(16x16,f32)  -> v_wmma_f32_16x16x4_f32
  return __builtin_amdgcn_wmma_f32_16x16x4_f32(false, a, false, b, (short)0, c,
                                               false, false);
}

// Async global->LDS 16B copy (per lane), tracked by ASYNCcnt.
__device__ __forceinline__ void async_ld_b128(uint32_t lds_off,
                                              const float* gsrc) {
  asm volatile("global_load_async_to_lds_b128 %0, %1, off"
               :
               : "v"(lds_off), "v"(gsrc)
               : "memory");
}
__device__ __forceinline__ void async_wait0() {
  asm volatile("s_wait_asynccnt 0x0" ::: "memory");
}

// ---------------- patch location table (computed analytically) ------------
__device__ __forceinline__ void patch_loc(int idx, float& y0, float& x0,
                                          float& y1, float& x1) {
  int level, base;
  if (idx < 1)       { level = 0; base = 0;  }
  else if (idx < 10) { level = 1; base = 1;  }
  else if (idx < 59) { level = 2; base = 10; }
  else               { level = 3; base = 59; }
  int n = (2 << level) - 1;               // 2^(level+1)-1
  float half = 1.0f / (float)(1 << level);
  int local = idx - base;
  int iy = local / n, ix = local % n;
  float cy = -1.0f + 2.0f * (float)(iy + 1) / (float)(n + 1);
  float cx = -1.0f + 2.0f * (float)(ix + 1) / (float)(n + 1);
  y0 = cy - half; x0 = cx - half; y1 = cy + half; x1 = cx + half;
}

// ---------------- integral image scans ----------------
__global__ __launch_bounds__(256) void row_scan_kernel(
    const float* __restrict__ img, float* __restrict__ intg) {
  __shared__ float buf[256];
  const int bid = blockIdx.x;           // n*224 + y
  const int n = bid / HH, y = bid % HH;
  const int tid = threadIdx.x;
  float v = (tid < WW) ? img[(size_t)n * HH * WW + (size_t)y * WW + tid] : 0.f;
  buf[tid] = v;
  __syncthreads();
  for (int off = 1; off < 256; off <<= 1) {
    float t = (tid >= off) ? buf[tid - off] : 0.f;
    __syncthreads();
    buf[tid] += t;
    __syncthreads();
  }
  float* row = intg + (size_t)n * ISZ + (size_t)(y + 1) * 225;
  if (tid < WW) row[tid + 1] = buf[tid];
  if (tid == 0) row[0] = 0.f;
}

__global__ __launch_bounds__(256) void col_scan_kernel(float* __restrict__ intg) {
  __shared__ float buf[256];
  const int bid = blockIdx.x;           // n*225 + x
  const int n = bid / 225, x = bid % 225;
  const int tid = threadIdx.x;
  float* col = intg + (size_t)n * ISZ + x;
  float v = (tid < HH) ? col[(size_t)(tid + 1) * 225] : 0.f;
  buf[tid] = v;
  __syncthreads();
  for (int off = 1; off < 256; off <<= 1) {
    float t = (tid >= off) ? buf[tid - off] : 0.f;
    __syncthreads();
    buf[tid] += t;
    __syncthreads();
  }
  if (tid < HH) col[(size_t)(tid + 1) * 225] = buf[tid];
  if (tid == 0) col[0] = 0.f;
}

// ---------------- patch extraction ----------------
__global__ __launch_bounds__(128) void patch_extract_kernel(
    const float* __restrict__ intg, const int* __restrict__ patch_idx,
    float* __restrict__ patches, float* __restrict__ pos3) {
  __shared__ float I[9][9];
  const int bid = blockIdx.x;           // token = n*16 + t
  const int n = bid >> 4;
  const int tid = threadIdx.x;
  float y0, x0, y1, x1;
  patch_loc(patch_idx[bid], y0, x0, y1, x1);
  const float* ig = intg + (size_t)n * ISZ;
  if (tid < 81) {
    int j = tid / 9, k2 = tid % 9;
    float ys = y0 + (y1 - y0) * ((float)j * 0.125f);
    float xs = x0 + (x1 - x0) * ((float)k2 * 0.125f);
    float py = fminf(fmaxf((ys + 1.f) * 0.5f * 224.f, 0.f), 224.f);
    float px = fminf(fmaxf((xs + 1.f) * 0.5f * 224.f, 0.f), 224.f);
    int iy0 = min(max((int)floorf(py), 0), 223);
    int ix0 = min(max((int)floorf(px), 0), 223);
    float wy = py - (float)iy0, wx = px - (float)ix0;
    float g00 = ig[iy0 * 225 + ix0];
    float g01 = ig[iy0 * 225 + ix0 + 1];
    float g10 = ig[(iy0 + 1) * 225 + ix0];
    float g11 = ig[(iy0 + 1) * 225 + ix0 + 1];
    I[j][k2] = g00 * (1.f - wy) * (1.f - wx) + g01 * (1.f - wy) * wx +
               g10 * wy * (1.f - wx) + g11 * wy * wx;
  }
  __syncthreads();
  if (tid < 64) {
    int a = tid >> 3, b = tid & 7;
    float s = I[a + 1][b + 1] - I[a][b + 1] - I[a + 1][b] + I[a][b];
    float area = ((y1 - y0) * 14.f) * ((x1 - x0) * 14.f);  // H/2/P = 14
    patches[(size_t)bid * 64 + tid] = s / area;
  }
  if (tid == 0) {
    pos3[bid * 3 + 0] = (y0 + y1) * 0.5f;
    pos3[bid * 3 + 1] = (x0 + x1) * 0.5f;
    pos3[bid * 3 + 2] = (y1 - y0) * 0.5f;
  }
}

// ---------------- fourier features + rope angles ----------------
__global__ __launch_bounds__(256) void posfeat_kernel(
    const float* __restrict__ pos3, const float* __restrict__ z4,
    const float* __restrict__ f4, const float* __restrict__ zr,
    const float* __restrict__ fr, float* __restrict__ four,
    float* __restrict__ cosb, float* __restrict__ sinb) {
  int gid = blockIdx.x * 256 + threadIdx.x;
  if (gid >= NT * NFREQS) return;
  int token = gid >> 6, j = gid & 63;
  float p0 = pos3[token * 3], p1 = pos3[token * 3 + 1], p2 = pos3[token * 3 + 2];
  float al = (z4[j * 3] * p0 + z4[j * 3 + 1] * p1 + z4[j * 3 + 2] * p2) *
             (1.f / 3.f);
  float th = f4[j] * al;
  four[(size_t)token * 128 + j] = sinf(th);
  four[(size_t)token * 128 + 64 + j] = cosf(th);
  if (j < 32) {
    float ar = (zr[j * 3] * p0 + zr[j * 3 + 1] * p1 + zr[j * 3 + 2] * p2) *
               (1.f / 3.f);
    float tr = fr[j] * ar;
    cosb[token * 32 + j] = cosf(tr);
    sinb[token * 32 + j] = sinf(tr);
  }
}

// ---------------- WMMA GEMM with async double-buffered LDS staging --------
// C[M x Ncols] (+)= op(A[M x K] @ B[K x Ncols])
// Block: 256 threads = 8 waves, tile 128 rows x 64 cols, K chunked by 32.
// Tiles stream in via global_load_async_to_lds_b128 (ASYNCcnt), double
// buffered so chunk c+1 copies overlap chunk c WMMAs.
// Row of A for output row r is (r*a_mul + a_off) (strided A for patch head).
// EPI: 0 = store, 1 = accumulate into C, 2 = relu(x)^2 store.
template <int EPI>
__global__ __launch_bounds__(256) void gemm_wmma(
    const float* __restrict__ A, int lda, int a_mul, int a_off,
    const float* __restrict__ B, int ldb, float* __restrict__ C, int ldc,
    int Ncols, int K) {
  // Row strides 36 / 68 floats: 16B-aligned rows (b128 async legal) and
  // odd-multiple-of-4 padding keeps A-column reads bank-conflict-free.
  __shared__ __align__(16) float As[2][128][36];
  __shared__ __align__(16) float Bs[2][32][68];
  const int tid = threadIdx.x;
  const int wave = tid >> 5;
  const int lane = tid & 31;
  const int hi = lane >> 4;          // wmma half (lanes 16-31)
  const int r = lane & 15;
  const int rowBase = blockIdx.y * 128;
  const int colBase = blockIdx.x * 64;
  const bool fullB = (colBase + 64 <= Ncols);  // uniform per block

  v8f acc[4] = {};

  auto stage = [&](int b, int k0) {
    // A tile: 128 x 32 floats = 1024 x b128
#pragma unroll
    for (int s2 = 0; s2 < 4; ++s2) {
      int idx4 = tid + s2 * 256;
      int rr = idx4 >> 3;
      int kk4 = (idx4 & 7) << 2;
      int grow = (rowBase + rr) * a_mul + a_off;
      uint32_t lds = (uint32_t)(uintptr_t)&As[b][rr][kk4];
      async_ld_b128(lds, A + (size_t)grow * lda + k0 + kk4);
    }
    if (fullB) {
      // B tile: 32 x 64 floats = 512 x b128
#pragma unroll
      for (int s2 = 0; s2 < 2; ++s2) {
        int idx4 = tid + s2 * 256;
        int kk = idx4 >> 4;
        int cc4 = (idx4 & 15) << 2;
        uint32_t lds = (uint32_t)(uintptr_t)&Bs[b][kk][cc4];
        async_ld_b128(lds, B + (size_t)(k0 + kk) * ldb + colBase + cc4);
      }
    } else {
      // guarded synchronous fallback (partial tile; zero-fill OOB columns)
      for (int i = tid; i < 32 * 64; i += 256) {
        int kk = i >> 6, cc = i & 63;
        int gc = colBase + cc;
        Bs[b][kk][cc] = (gc < Ncols) ? B[(size_t)(k0 + kk) * ldb + gc] : 0.f;
      }
    }
  };

  const int nchunks = K >> 5;
  stage(0, 0);
  async_wait0();
  __syncthreads();

  for (int c = 0; c < nchunks; ++c) {
    const int cur = c & 1;
    if (c + 1 < nchunks) stage(cur ^ 1, (c + 1) << 5);
#pragma unroll
    for (int kk = 0; kk < 32; kk += 4) {
      v2f a;
      a.x = As[cur][wave * 16 + r][kk + 2 * hi];
      a.y = As[cur][wave * 16 + r][kk + 2 * hi + 1];
#pragma unroll
      for (int j = 0; j < 4; ++j) {
        v2f b;
        b.x = Bs[cur][kk + 2 * hi][j * 16 + r];
        b.y = Bs[cur][kk + 2 * hi + 1][j * 16 + r];
        acc[j] = wmma4(a, b, acc[j]);
      }
    }
    async_wait0();
    __syncthreads();
  }

#pragma unroll
  for (int j = 0; j < 4; ++j) {
    int col = colBase + j * 16 + r;
    if (col < Ncols) {
#pragma unroll
      for (int i = 0; i < 8; ++i) {
        int row = rowBase + wave * 16 + i + 8 * hi;
        size_t off = (size_t)row * ldc + col;
        float v = acc[j][i];
        if (EPI == 1) {
          C[off] += v;
        } else if (EPI == 2) {
          float t = fmaxf(v, 0.f);
          C[off] = t * t;
        } else {
          C[off] = v;
        }
      }
    }
  }
}

// ---------------- rmsnorm ----------------
__global__ __launch_bounds__(128) void rmsnorm_kernel(
    const float* __restrict__ h, float* __restrict__ hn) {
  __shared__ float partial[4];
  const int token = blockIdx.x, tid = threadIdx.x;
  const float* hp = h + (size_t)token * DIM;
  float x0 = hp[tid], x1 = hp[tid + 128], x2 = hp[tid + 256];
  float ss = x0 * x0 + x1 * x1 + x2 * x2;
  for (int m = 1; m < 32; m <<= 1) ss += __shfl_xor(ss, m, 32);
  if ((tid & 31) == 0) partial[tid >> 5] = ss;
  __syncthreads();
  float tot = partial[0] + partial[1] + partial[2] + partial[3];
  float rs = rsqrtf(tot * (1.f / (float)DIM) + 1e-6f);
  float* hq = hn + (size_t)token * DIM;
  hq[tid] = x0 * rs;
  hq[tid + 128] = x1 * rs;
  hq[tid + 256] = x2 * rs;
}

// ---------------- rope (in-place on q,k inside qkv) ----------------
__global__ __launch_bounds__(256) void rope_kernel(
    float* __restrict__ qkv, const float* __restrict__ cosb,
    const float* __restrict__ sinb) {
  int gid = blockIdx.x * 256 + threadIdx.x;
  if (gid >= NT * NHEADS * 32) return;
  int d = gid & 31;
  int head = (gid >> 5) % NHEADS;
  int token = gid / (NHEADS * 32);
  float c = cosb[token * 32 + d], s = sinb[token * 32 + d];
  float* q = qkv + (size_t)token * (3 * DIM) + head * HD;
  float q1 = q[d], q2 = q[d + 32];
  q[d] = q1 * c - q2 * s;
  q[d + 32] = q1 * s + q2 * c;
  float* k = q + DIM;
  float k1 = k[d], k2 = k[d + 32];
  k[d] = k1 * c - k2 * s;
  k[d + 32] = k1 * s + k2 * c;
}

// ---------------- attention: one wave per (image, head) ----------------
__global__ __launch_bounds__(256) void attn_kernel(
    const float* __restrict__ qkv, float* __restrict__ o) {
  __shared__ float Pl[8][16][17];
  const int tid = threadIdx.x, wave = tid >> 5, lane = tid & 31;
  const int hi = lane >> 4, r = lane & 15;
  const int item = blockIdx.x * 8 + wave;  // n*NHEADS + head
  const int n = item / NHEADS, head = item % NHEADS;
  const float* base = qkv + (size_t)(n * TT) * (3 * DIM) + head * HD;

  // S = Q K^T  (16x16, K-dim = 64)
  v8f s = {};
#pragma unroll
  for (int kk = 0; kk < HD; kk += 4) {
    v2f a, b;
    const float* qrow = base + (size_t)r * (3 * DIM) + kk + 2 * hi;
    a.x = qrow[0];
    a.y = qrow[1];
    const float* krow = base + DIM + (size_t)r * (3 * DIM) + kk + 2 * hi;
    b.x = krow[0];
    b.y = krow[1];
    s = wmma4(a, b, s);
  }
  // causal softmax along keys (row = query = i + 8*hi, col = key = r)
#pragma unroll
  for (int i = 0; i < 8; ++i) {
    int M = i + 8 * hi;
    float v = (r <= M) ? s[i] * 0.125f : -1e30f;
    float mx = v;
    for (int m = 1; m < 16; m <<= 1) mx = fmaxf(mx, __shfl_xor(mx, m, 32));
    float p = (r <= M) ? expf(v - mx) : 0.f;
    float sum = p;
    for (int m = 1; m < 16; m <<= 1) sum += __shfl_xor(sum, m, 32);
    Pl[wave][M][r] = p / sum;
  }
  __syncthreads();
  // O = P V  (16x64 via 4 tiles, K-dim = 16 tokens)
  v8f accO[4] = {};
#pragma unroll
  for (int j = 0; j < 4; ++j) {
#pragma unroll
    for (int kk = 0; kk < 16; kk += 4) {
      v2f a, b;
      a.x = Pl[wave][r][kk + 2 * hi];
      a.y = Pl[wave][r][kk + 2 * hi + 1];
      const float* vrow0 =
          base + 2 * DIM + (size_t)(kk + 2 * hi) * (3 * DIM) + j * 16 + r;
      b.x = vrow0[0];
      b.y = vrow0[3 * DIM];
      accO[j] = wmma4(a, b, accO[j]);
    }
  }
#pragma unroll
  for (int j = 0; j < 4; ++j) {
#pragma unroll
    for (int i = 0; i < 8; ++i) {
      int tok = n * TT + i + 8 * hi;
      o[(size_t)tok * DIM + head * HD + j * 16 + r] = accO[j][i];
    }
  }
}

// ---------------- class head (small N=10) ----------------
__global__ __launch_bounds__(256) void classhead_kernel(
    const float* __restrict__ hn, const float* __restrict__ Wout,
    float* __restrict__ out) {
  int gid = blockIdx.x * 256 + threadIdx.x;  // NT * 16
  int c = gid & 15, token = gid >> 4;
  if (token >= NT || c >= 10) return;
  const float* hp = hn + (size_t)token * DIM;
  float sum = 0.f;
  for (int k = 0; k < DIM; ++k) sum += hp[k] * Wout[k * 10 + c];
  out[token * 10 + c] = sum;
}

// ---------------- launcher ----------------
extern "C" void kernel_launch(void* const* d_in, const int* in_sizes, int n_in,
                              void* d_out, int out_size, void* d_ws,
                              size_t ws_size, hipStream_t stream) {
  (void)in_sizes; (void)n_in; (void)out_size; (void)ws_size;
  const float* images     = (const float*)d_in[0];
  const int*   patch_idx  = (const int*)d_in[1];
  const float* W_patchify = (const float*)d_in[2];
  const float* z_four     = (const float*)d_in[3];
  const float* f_four     = (const float*)d_in[4];
  const float* W_emb      = (const float*)d_in[5];
  const float* Wqkv       = (const float*)d_in[6];
  const float* Wo         = (const float*)d_in[7];
  const float* W1         = (const float*)d_in[8];
  const float* W2         = (const float*)d_in[9];
  const float* z_rope     = (const float*)d_in[10];
  const float* f_rope     = (const float*)d_in[11];
  const float* W_out      = (const float*)d_in[12];
  const float* W_patch_head = (const float*)d_in[13];
  float* out = (float*)d_out;
  float* ws = (float*)d_ws;

  // workspace layout (floats)
  float* intg    = ws;                                   // 1024*50625
  float* patches = ws + (size_t)NIMG * ISZ;              // NT*64
  float* pos3    = patches + (size_t)NT * 64;            // NT*3
  float* cosb    = pos3 + (size_t)NT * 3;                // NT*32
  float* sinb    = cosb + (size_t)NT * 32;               // NT*32
  float* four    = sinb + (size_t)NT * 32;               // NT*128
  // activation buffers overlap the integral region (dead after extraction)
  float* h    = ws;                                      // NT*384
  float* hn   = h + (size_t)NT * DIM;                    // NT*384
  float* qkv  = hn + (size_t)NT * DIM;                   // NT*1152
  float* obuf = qkv + (size_t)NT * 3 * DIM;              // NT*384
  float* mid  = obuf + (size_t)NT * DIM;                 // NT*768 (fits < intg)

  // 1) integral images
  row_scan_kernel<<<NIMG * HH, 256, 0, stream>>>(images, intg);
  col_scan_kernel<<<NIMG * 225, 256, 0, stream>>>(intg);
  // 2) patch extraction + positions
  patch_extract_kernel<<<NT, 128, 0, stream>>>(intg, patch_idx, patches, pos3);
  // 3) fourier features + rope angles
  posfeat_kernel<<<(NT * NFREQS) / 256, 256, 0, stream>>>(
      pos3, z_four, f_four, z_rope, f_rope, four, cosb, sinb);
  // 4) h = patches @ W_patchify ; h += four @ W_emb
  gemm_wmma<0><<<dim3(6, NT / 128), 256, 0, stream>>>(
      patches, 64, 1, 0, W_patchify, DIM, h, DIM, DIM, 64);
  gemm_wmma<1><<<dim3(6, NT / 128), 256, 0, stream>>>(
      four, 128, 1, 0, W_emb, DIM, h, DIM, DIM, 128);

  // 5) transformer layers
  for (int l = 0; l < DEPTH; ++l) {
    rmsnorm_kernel<<<NT, 128, 0, stream>>>(h, hn);
    gemm_wmma<0><<<dim3(18, NT / 128), 256, 0, stream>>>(
        hn, DIM, 1, 0, Wqkv + (size_t)l * DIM * 3 * DIM, 3 * DIM, qkv, 3 * DIM,
        3 * DIM, DIM);
    rope_kernel<<<(NT * NHEADS * 32) / 256, 256, 0, stream>>>(qkv, cosb, sinb);
    attn_kernel<<<(NIMG * NHEADS) / 8, 256, 0, stream>>>(qkv, obuf);
    gemm_wmma<1><<<dim3(6, NT / 128), 256, 0, stream>>>(
        obuf, DIM, 1, 0, Wo + (size_t)l * DIM * DIM, DIM, h, DIM, DIM, DIM);
    rmsnorm_kernel<<<NT, 128, 0, stream>>>(h, hn);
    gemm_wmma<2><<<dim3(12, NT / 128), 256, 0, stream>>>(
        hn, DIM, 1, 0, W1 + (size_t)l * DIM * MLPD, MLPD, mid, MLPD, MLPD, DIM);
    gemm_wmma<1><<<dim3(6, NT / 128), 256, 0, stream>>>(
        mid, MLPD, 1, 0, W2 + (size_t)l * MLPD * DIM, DIM, h, DIM, DIM, MLPD);
  }

  // 6) final norm + heads
  rmsnorm_kernel<<<NT, 128, 0, stream>>>(h, hn);
  classhead_kernel<<<(NT * 16) / 256, 256, 0, stream>>>(hn, W_out, out);
  // patch_logits = hf[:, -1] @ W_patch_head : A row n -> token n*16+15
  gemm_wmma<0><<<dim3(5, NIMG / 128), 256, 0, stream>>>(
      hn, DIM, TT, TT - 1, W_patch_head, LPATCH, out + (size_t)NT * 10, LPATCH,
      LPATCH, DIM);
}